// BlockSSM_39857296507005
// MI455X (gfx1250) — compile-verified
//
#include <hip/hip_runtime.h>

typedef __attribute__((ext_vector_type(2))) float v2f;
typedef __attribute__((ext_vector_type(8))) float v8f;

#define NX 128
#define NY 32
#define NU 32
#define ND 16
#define BATCH 256
#define NSTEPS 2048

// D = A(16x4, f32) * B(4x16, f32) + C(16x16, f32)
static __device__ __forceinline__ v8f wmma_f32(v2f a, v2f b, v8f c) {
  return __builtin_amdgcn_wmma_f32_16x16x4_f32(false, a, false, b, (short)0, c,
                                               false, false);
}

// A-fragment from global row-major A[*, lda], tile origin (m0, k0).
// Lanes 0-15: (M=lane, K=k0+0..1); lanes 16-31: (M=lane-16, K=k0+2..3).
static __device__ __forceinline__ v2f a_frag_g(const float* A, int lda, int m0, int k0) {
  const int lane = threadIdx.x & 31;
  const float* p = A + (size_t)(m0 + (lane & 15)) * lda + k0 + ((lane >> 4) << 1);
  return v2f{p[0], p[1]};
}

// B-fragment of W^T where W is [N, K] row-major: B[k][n] = W[n][k].
// Lanes 0-15 hold (K=k0+0..1, N=n0+lane); lanes 16-31 hold (K=k0+2..3, N=n0+lane-16).
static __device__ __forceinline__ v2f b_frag_wT(const float* W, int ldw, int n0, int k0) {
  const int lane = threadIdx.x & 31;
  const float* p = W + (size_t)(n0 + (lane & 15)) * ldw + k0 + ((lane >> 4) << 1);
  return v2f{p[0], p[1]};
}

// ---------------------------------------------------------------------------
// Kernel 1: FU = Uf @ Wu^T + bu ; FD = Df @ Wd^T + bd  (fully parallel GEMMs)
// Block = 256 threads (8 waves); block owns a 16-row slab of [NSTEPS*BATCH];
// wave w owns N-tile w of NX=128.
// ---------------------------------------------------------------------------
__global__ __launch_bounds__(256) void fufd_kernel(
    const float* __restrict__ Uf, const float* __restrict__ Df,
    const float* __restrict__ Wu, const float* __restrict__ bu,
    const float* __restrict__ Wd, const float* __restrict__ bd,
    float* __restrict__ FU, float* __restrict__ FD) {
  const int lane = threadIdx.x & 31;
  const int wave = threadIdx.x >> 5;
  const int m0 = blockIdx.x << 4;
  const int n0 = wave << 4;

  v8f accU0 = {}, accU1 = {};
  v8f accD0 = {}, accD1 = {};
#pragma unroll
  for (int k0 = 0; k0 < NU; k0 += 8) {
    accU0 = wmma_f32(a_frag_g(Uf, NU, m0, k0), b_frag_wT(Wu, NU, n0, k0), accU0);
    accU1 = wmma_f32(a_frag_g(Uf, NU, m0, k0 + 4), b_frag_wT(Wu, NU, n0, k0 + 4), accU1);
  }
#pragma unroll
  for (int k0 = 0; k0 < ND; k0 += 8) {
    accD0 = wmma_f32(a_frag_g(Df, ND, m0, k0), b_frag_wT(Wd, ND, n0, k0), accD0);
    accD1 = wmma_f32(a_frag_g(Df, ND, m0, k0 + 4), b_frag_wT(Wd, ND, n0, k0 + 4), accD1);
  }
  const v8f accU = accU0 + accU1;
  const v8f accD = accD0 + accD1;

  const int n = n0 + (lane & 15);
  const int mb = (lane >> 4) << 3;  // 0 or 8
  const float bun = bu[n];
  const float bdn = bd[n];
#pragma unroll
  for (int j = 0; j < 8; ++j) {
    const size_t idx = (size_t)(m0 + mb + j) * NX + n;
    FU[idx] = accU[j] + bun;
    FD[idx] = accD[j] + bdn;
  }
}

// ---------------------------------------------------------------------------
// Kernel 2: the serial scan (recurrence only -- Y is deferred to kernel 3).
// Batch rows are independent across steps: each of 16 blocks owns a 16-row
// batch slab. x ping-pongs through LDS (pitch 130 floats: conflict-free,
// 8B-aligned column reads). Wave w keeps Wx^T fragments for its N-tile in
// 64 VGPRs. 4 accumulators cut the WMMA dependence chain from 32 to 8.
// ---------------------------------------------------------------------------
__global__ __launch_bounds__(256) void scan_kernel(
    const float* __restrict__ x0,
    const float* __restrict__ Wx, const float* __restrict__ bx,
    const float* __restrict__ FU, const float* __restrict__ FD,
    float* __restrict__ X) {
  __shared__ float xbuf[2][16][NX + 2];

  const int lane = threadIdx.x & 31;
  const int wave = threadIdx.x >> 5;
  const int b0 = blockIdx.x << 4;     // batch-row slab
  const int n0 = wave << 4;           // this wave's N-tile in NX
  const int nl = lane & 15;
  const int koff = (lane >> 4) << 1;  // 0 or 2
  const int mb = (lane >> 4) << 3;    // 0 or 8

  // Loop-invariant Wx^T fragments (registers, 64 VGPRs).
  v2f wxf[32];
#pragma unroll
  for (int kk = 0; kk < 32; ++kk) wxf[kk] = b_frag_wT(Wx, NX, n0, kk * 4);

  const float bxn = bx[n0 + nl];

  // Seed state with x0 slab.
  for (int i = threadIdx.x; i < 16 * NX; i += 256) {
    const int r = i >> 7, c = i & (NX - 1);
    xbuf[0][r][c] = x0[(size_t)(b0 + r) * NX + c];
  }
  __syncthreads();

  for (int t = 0; t < NSTEPS; ++t) {
    const int p = t & 1;
    const size_t baseX = (size_t)t * BATCH * NX;

    // Issue this step's FU/FD loads first so they overlap the WMMA chain.
    float fu_v[8], fd_v[8];
#pragma unroll
    for (int j = 0; j < 8; ++j) {
      const size_t idx = baseX + (size_t)(b0 + mb + j) * NX + (n0 + nl);
      fu_v[j] = FU[idx];
      fd_v[j] = FD[idx];
    }

    // acc = x_t @ Wx^T (16x16 tile, K=128) with 4 independent accumulators.
    v8f a0 = {}, a1 = {}, a2 = {}, a3 = {};
#pragma unroll
    for (int kk = 0; kk < 8; ++kk) {
      {
        const int k = kk * 4 + koff;
        v2f a = {xbuf[p][nl][k], xbuf[p][nl][k + 1]};
        a0 = wmma_f32(a, wxf[kk], a0);
      }
      {
        const int k = (kk + 8) * 4 + koff;
        v2f a = {xbuf[p][nl][k], xbuf[p][nl][k + 1]};
        a1 = wmma_f32(a, wxf[kk + 8], a1);
      }
      {
        const int k = (kk + 16) * 4 + koff;
        v2f a = {xbuf[p][nl][k], xbuf[p][nl][k + 1]};
        a2 = wmma_f32(a, wxf[kk + 16], a2);
      }
      {
        const int k = (kk + 24) * 4 + koff;
        v2f a = {xbuf[p][nl][k], xbuf[p][nl][k + 1]};
        a3 = wmma_f32(a, wxf[kk + 24], a3);
      }
    }
    const v8f acc = (a0 + a1) + (a2 + a3);

    // x_{t+1} = 2*(acc + bx + fu_t) + fd_t ; write global X + LDS ping-pong.
#pragma unroll
    for (int j = 0; j < 8; ++j) {
      const size_t idx = baseX + (size_t)(b0 + mb + j) * NX + (n0 + nl);
      const float v = 2.0f * (acc[j] + bxn + fu_v[j]) + fd_v[j];
      X[idx] = v;
      xbuf[1 - p][mb + j][n0 + nl] = v;
    }

    // Warm next step's FU/FD cachelines (global_prefetch_b8, no counters).
    if (t + 1 < NSTEPS) {
      const size_t nb = baseX + (size_t)BATCH * NX + (n0 + nl);
#pragma unroll
      for (int j = 0; j < 8; ++j) {
        const size_t nidx = nb + (size_t)(b0 + mb + j) * NX;
        __builtin_prefetch(FU + nidx, 0, 0);
        __builtin_prefetch(FD + nidx, 0, 0);
      }
    }

    __syncthreads();  // one barrier/step is sufficient with ping-pong buffers
  }
}

// ---------------------------------------------------------------------------
// Kernel 3: Y = X @ Wy^T + by, fully parallel (X is complete after the scan).
// Block = 256 threads (8 waves) covers a 64-row slab of [NSTEPS*BATCH]:
// wave w -> rows (w>>1)*16, N-tile (w&1) of NY=32.
// ---------------------------------------------------------------------------
__global__ __launch_bounds__(256) void y_kernel(
    const float* __restrict__ X, const float* __restrict__ Wy,
    const float* __restrict__ by, float* __restrict__ Y) {
  const int lane = threadIdx.x & 31;
  const int wave = threadIdx.x >> 5;
  const int m0 = (blockIdx.x << 6) + ((wave >> 1) << 4);
  const int n0 = (wave & 1) << 4;

  v8f a0 = {}, a1 = {}, a2 = {}, a3 = {};
#pragma unroll
  for (int kk = 0; kk < 8; ++kk) {
    a0 = wmma_f32(a_frag_g(X, NX, m0, kk * 4),
                  b_frag_wT(Wy, NX, n0, kk * 4), a0);
    a1 = wmma_f32(a_frag_g(X, NX, m0, (kk + 8) * 4),
                  b_frag_wT(Wy, NX, n0, (kk + 8) * 4), a1);
    a2 = wmma_f32(a_frag_g(X, NX, m0, (kk + 16) * 4),
                  b_frag_wT(Wy, NX, n0, (kk + 16) * 4), a2);
    a3 = wmma_f32(a_frag_g(X, NX, m0, (kk + 24) * 4),
                  b_frag_wT(Wy, NX, n0, (kk + 24) * 4), a3);
  }
  const v8f acc = (a0 + a1) + (a2 + a3);

  const int n = n0 + (lane & 15);
  const int mb = (lane >> 4) << 3;
  const float byn = by[n];
#pragma unroll
  for (int j = 0; j < 8; ++j)
    Y[(size_t)(m0 + mb + j) * NY + n] = acc[j] + byn;
}

extern "C" void kernel_launch(void* const* d_in, const int* in_sizes, int n_in,
                              void* d_out, int out_size, void* d_ws, size_t ws_size,
                              hipStream_t stream) {
  (void)in_sizes; (void)n_in; (void)out_size; (void)d_ws; (void)ws_size;

  const float* x0 = (const float*)d_in[0];
  // d_in[1] = Yf (unused by the reference computation)
  const float* Uf = (const float*)d_in[2];
  const float* Df = (const float*)d_in[3];
  const float* Wx = (const float*)d_in[4];
  const float* bx = (const float*)d_in[5];
  const float* Wu = (const float*)d_in[6];
  const float* bu = (const float*)d_in[7];
  const float* Wd = (const float*)d_in[8];
  const float* bd = (const float*)d_in[9];
  const float* Wy = (const float*)d_in[10];
  const float* by = (const float*)d_in[11];

  float* X  = (float*)d_out;                     // [NSTEPS,BATCH,NX]
  float* Y  = X + (size_t)NSTEPS * BATCH * NX;   // [NSTEPS,BATCH,NY]
  float* FU = Y + (size_t)NSTEPS * BATCH * NY;   // [NSTEPS,BATCH,NX]
  float* FD = FU + (size_t)NSTEPS * BATCH * NX;  // [NSTEPS,BATCH,NX]

  fufd_kernel<<<(NSTEPS * BATCH) / 16, 256, 0, stream>>>(Uf, Df, Wu, bu, Wd, bd,
                                                         FU, FD);
  scan_kernel<<<BATCH / 16, 256, 0, stream>>>(x0, Wx, bx, FU, FD, X);
  y_kernel<<<(NSTEPS * BATCH) / 64, 256, 0, stream>>>(X, Wy, by, Y);
}